// F2GNN_64055142252772
// MI455X (gfx1250) — compile-verified
//
#include <hip/hip_runtime.h>

typedef __attribute__((ext_vector_type(2))) float v2f;
typedef __attribute__((ext_vector_type(8))) float v8f;

#define IN_DIM 128
#define FEAT   64
#define EPS    0.5f

// ---------------------------------------------------------------------------
// small utility kernels
// ---------------------------------------------------------------------------
__global__ void zero_f32(float* __restrict__ p, int n) {
    int i = blockIdx.x * blockDim.x + threadIdx.x;
    if (i < n) p[i] = 0.0f;
}

__global__ void deg_kernel(const long long* __restrict__ dst, float* __restrict__ deg, int E) {
    int e = blockIdx.x * blockDim.x + threadIdx.x;
    if (e < E) atomicAdd(&deg[(int)dst[e]], 1.0f);
}

__global__ void norm_kernel(float* __restrict__ deg, int n) {
    int i = blockIdx.x * blockDim.x + threadIdx.x;
    if (i < n) deg[i] = rsqrtf(fmaxf(deg[i], 1.0f));
}

// ---------------------------------------------------------------------------
// fp32 WMMA GEMM: out[nrows x 64] = act(A[nrows x K] @ W[K x 64] + bias)
//
// - One wave computes a full 16-row x 64-col tile (4 v8f accumulators) so the
//   A tile is loaded once per wave (global_load_b64 per k-chunk).
// - B panel (K x 64, <=32KB) is staged once per block into LDS in a k-pair
//   interleaved layout so each inner-loop fetch is a single ds_load_b64:
//       ldsB[(k>>1)*128 + n*2 + (k&1)] = W[k][n]
// - A rows are clamped (not predicated): garbage rows only feed C rows that
//   the guarded store discards, so no exec-mask churn around loads.
//
// ISA layouts (CDNA5 ISA §7.12.2, 32-bit):
//   A 16x4 : lanes 0-15 -> M=lane, v0=K0,v1=K1 ; lanes 16-31 -> M=lane-16, v0=K2,v1=K3
//   B 4x16 : lanes 0-15 -> N=lane, v0=K0,v1=K1 ; lanes 16-31 -> N=lane-16, v0=K2,v1=K3
//   C 16x16: VGPR r: lanes 0-15 -> M=r, N=lane ; lanes 16-31 -> M=8+r, N=lane-16
// ---------------------------------------------------------------------------
template <int K, bool RELU, bool WRITE_ACC>
__global__ void gemm_wmma_f32(const float* __restrict__ A,
                              const float* __restrict__ W,
                              const float* __restrict__ bias,
                              float* __restrict__ out,
                              float* __restrict__ acc,
                              int nrows) {
    __shared__ float ldsB[K * FEAT];

    // cooperative stage + relayout of the weight panel
    for (int i = threadIdx.x; i < K * FEAT; i += blockDim.x) {
        int k = i >> 6;       // row of W
        int n = i & 63;       // col of W
        ldsB[(k >> 1) * 128 + n * 2 + (k & 1)] = W[i];
    }
    __syncthreads();

    int wave = (blockIdx.x * blockDim.x + threadIdx.x) >> 5;
    int lane = threadIdx.x & 31;
    int m0 = wave * 16;
    if (m0 >= nrows) return;                 // wave-uniform, EXEC stays all-1s

    int mr = lane & 15;   // row within tile (A) / col within 16-col tile (B, C)
    int kh = lane >> 4;   // K-half select: 0 -> K0/K1, 1 -> K2/K3

    int arow = m0 + mr;
    if (arow >= nrows) arow = nrows - 1;     // clamp: dead rows never stored
    const v2f* arow2 = (const v2f*)(A + (size_t)arow * K);
    const v2f* ldsB2 = (const v2f*)ldsB;

    v8f c0 = {}, c1 = {}, c2 = {}, c3 = {};

#pragma unroll 4
    for (int k0 = 0; k0 < K; k0 += 4) {
        int kp = (k0 >> 1) + kh;             // k-pair index for this half-wave
        v2f a = arow2[kp];

        const v2f* bp = ldsB2 + kp * 64;
        v2f b0 = bp[mr];
        v2f b1 = bp[16 + mr];
        v2f b2 = bp[32 + mr];
        v2f b3 = bp[48 + mr];

        c0 = __builtin_amdgcn_wmma_f32_16x16x4_f32(false, a, false, b0, (short)0, c0, false, false);
        c1 = __builtin_amdgcn_wmma_f32_16x16x4_f32(false, a, false, b1, (short)0, c1, false, false);
        c2 = __builtin_amdgcn_wmma_f32_16x16x4_f32(false, a, false, b2, (short)0, c2, false, false);
        c3 = __builtin_amdgcn_wmma_f32_16x16x4_f32(false, a, false, b3, (short)0, c3, false, false);
    }

    float bb0 = bias[mr], bb1 = bias[16 + mr], bb2 = bias[32 + mr], bb3 = bias[48 + mr];

#pragma unroll
    for (int r = 0; r < 8; ++r) {
        int m = m0 + kh * 8 + r;
        if (m < nrows) {
            size_t off = (size_t)m * FEAT + mr;
            float v0 = c0[r] + bb0;
            float v1 = c1[r] + bb1;
            float v2 = c2[r] + bb2;
            float v3 = c3[r] + bb3;
            if (RELU) {
                v0 = fmaxf(v0, 0.0f); v1 = fmaxf(v1, 0.0f);
                v2 = fmaxf(v2, 0.0f); v3 = fmaxf(v3, 0.0f);
            }
            out[off]      = v0;
            out[off + 16] = v1;
            out[off + 32] = v2;
            out[off + 48] = v3;
            if (WRITE_ACC) {
                acc[off]      = EPS * v0;
                acc[off + 16] = EPS * v1;
                acc[off + 32] = EPS * v2;
                acc[off + 48] = EPS * v3;
            }
        }
    }
}

// ---------------------------------------------------------------------------
// Edge kernel: 16 lanes per edge, each lane owns a float4 feature slice.
// Per head h (= lane16/4): s = tanh(sum_d hd[d]*gW[d] + hs[d]*gW[16+d] + gb)
// reduced over the aligned 4-lane group with shfl_xor (wave32).
// Scatter: acc[dst] += hs * s * norm[dst] * norm[src]  (global f32 atomics,
// L2-resident since acc fits in the 192MB L2).
// Only layer 1's gate weights matter (layer 0 is dead in the reference).
// ---------------------------------------------------------------------------
__global__ void edge_kernel(const float* __restrict__ x,
                            const long long* __restrict__ src,
                            const long long* __restrict__ dst,
                            const float* __restrict__ norm,
                            const float* __restrict__ gW,   // 32 floats (layer 1)
                            const float* __restrict__ gB,   // 1 float   (layer 1)
                            float* __restrict__ acc,
                            int E) {
    int gtid = blockIdx.x * blockDim.x + threadIdx.x;
    int e = gtid >> 4;
    if (e >= E) return;
    int l16 = gtid & 15;

    int s = (int)src[e];
    int d = (int)dst[e];

    const float4 hs = *(const float4*)(x + (size_t)s * FEAT + l16 * 4);
    const float4 hd = *(const float4*)(x + (size_t)d * FEAT + l16 * 4);

    int doff = (l16 & 3) * 4;   // position within the head's 16-dim slice
    float p = hd.x * gW[doff]      + hd.y * gW[doff + 1]
            + hd.z * gW[doff + 2]  + hd.w * gW[doff + 3]
            + hs.x * gW[16 + doff]     + hs.y * gW[16 + doff + 1]
            + hs.z * gW[16 + doff + 2] + hs.w * gW[16 + doff + 3];

    // reduce the 4 aligned lanes of this head
    p += __shfl_xor(p, 1, 32);
    p += __shfl_xor(p, 2, 32);

    float sg = tanhf(p + gB[0]) * norm[d] * norm[s];

    float* ap = acc + (size_t)d * FEAT + l16 * 4;
    atomicAdd(ap + 0, hs.x * sg);
    atomicAdd(ap + 1, hs.y * sg);
    atomicAdd(ap + 2, hs.z * sg);
    atomicAdd(ap + 3, hs.w * sg);
}

// ---------------------------------------------------------------------------
extern "C" void kernel_launch(void* const* d_in, const int* in_sizes, int n_in,
                              void* d_out, int out_size, void* d_ws, size_t ws_size,
                              hipStream_t stream) {
    const float*     h   = (const float*)d_in[0];
    const long long* src = (const long long*)d_in[1];
    const long long* dst = (const long long*)d_in[2];
    const float*     t1W = (const float*)d_in[3];
    const float*     t1b = (const float*)d_in[4];
    const float*     t2W = (const float*)d_in[5];
    const float*     t2b = (const float*)d_in[6];
    const float*     gW  = (const float*)d_in[7];   // [2, 32, 1]
    const float*     gB  = (const float*)d_in[8];   // [2, 1]
    float* out = (float*)d_out;

    int N = in_sizes[0] / IN_DIM;
    int E = in_sizes[1];

    float* ws  = (float*)d_ws;
    float* x   = ws;                          // [N,64]
    float* acc = ws + (size_t)N * FEAT;       // [N,64]
    float* deg = ws + 2 * (size_t)N * FEAT;   // [N]  (deg, then norm in place)

    // 1) in-degree -> norm
    zero_f32<<<(N + 255) / 256, 256, 0, stream>>>(deg, N);
    deg_kernel<<<(E + 255) / 256, 256, 0, stream>>>(dst, deg, E);
    norm_kernel<<<(N + 255) / 256, 256, 0, stream>>>(deg, N);

    // 2) x = relu(h @ t1_W + t1_b); acc = EPS * x   (fp32 WMMA)
    int mtiles = (N + 15) / 16;
    int gblocks = (mtiles + 7) / 8;           // 8 waves (256 threads) per block
    gemm_wmma_f32<IN_DIM, true, true><<<gblocks, 256, 0, stream>>>(h, t1W, t1b, x, acc, N);

    // 3) edge pass (only layer 1 of the reference is live)
    long long ethreads = (long long)E * 16;
    edge_kernel<<<(int)((ethreads + 255) / 256), 256, 0, stream>>>(
        x, src, dst, deg, gW + 32, gB + 1, acc, E);

    // 4) out = acc @ t2_W + t2_b   (fp32 WMMA)
    gemm_wmma_f32<FEAT, false, false><<<gblocks, 256, 0, stream>>>(acc, t2W, t2b, out, nullptr, N);
}